// MyLSTM_1314259993205
// MI455X (gfx1250) — compile-verified
//
#include <hip/hip_runtime.h>
#include <cstdint>
#include <cstddef>

// ---------------------------------------------------------------------------
// Bidirectional 3-layer LSTM for MI455X (gfx1250), fp32 via V_WMMA_F32_16X16X4_F32.
//   Phase A (per layer, per dir): gatesPre = X(32768x800) @ W_ih^T + b   (WMMA GEMM)
//   Phase B (per layer, t-serial): gates = h @ W_hh^T + gatesPre; LSTM cell fused.
// Workspace use: 2*TB*1600 + TB*800 + 8*B*H floats ~= 528 MB.
// ---------------------------------------------------------------------------

typedef __attribute__((ext_vector_type(2))) float v2f;
typedef __attribute__((ext_vector_type(8))) float v8f;

#define T_SEQ   128
#define BATCH   256
#define IN_DIM  800
#define HID     400
#define GATES   1600   // 4*HID
#define LAYERS  3

#define LDPAD   20     // padded LDS row (floats); keeps {k,k+1} pairs 8B aligned

__device__ __forceinline__ v2f ld2s(const float* p) {
  v2f r; r.x = p[0]; r.y = p[1]; return r;
}

__device__ __forceinline__ float fast_sigmoid(float x) {
  return __builtin_amdgcn_rcpf(1.0f + __expf(-x));
}
__device__ __forceinline__ float fast_tanh(float x) {
  // tanh(x) = 1 - 2/(e^{2x}+1); saturates correctly for large |x|
  return 1.0f - 2.0f * __builtin_amdgcn_rcpf(1.0f + __expf(2.0f * x));
}

__global__ void zero_f32(float* __restrict__ p, int n) {
  int i = blockIdx.x * blockDim.x + threadIdx.x;
  if (i < n) p[i] = 0.0f;
}

// ---------------------------------------------------------------------------
// Phase A: out[M=32768][N=1600] = X[M][800] * W[N][800]^T + bias[N]
// Block tile 128x64, BK=16. 256 threads = 8 waves; wave = 32x32 (2x2 WMMA tiles).
// ---------------------------------------------------------------------------
__global__ __launch_bounds__(256) void gemm_gates(
    const float* __restrict__ X, const float* __restrict__ W,
    const float* __restrict__ bias, float* __restrict__ out)
{
  __shared__ float As[2][128][LDPAD];
  __shared__ float Bs[2][64][LDPAD];

  const int tid  = threadIdx.x;
  const int lane = tid & 31;
  const int wave = tid >> 5;
  const int ln   = lane & 15;
  const int klo  = (lane < 16) ? 0 : 2;   // A/B fragment k-pair base
  const int hi8  = (lane < 16) ? 0 : 8;   // C/D row offset

  const int mBase = blockIdx.x * 128;
  const int nBase = blockIdx.y * 64;
  const int wm = (wave & 3) * 32;
  const int wn = (wave >> 2) * 32;

  // Load index mapping (A: 2 float4/thread, B: 1 float4/thread)
  const int ar = tid >> 1;               // 0..127
  const int ac = (tid & 1) * 8;          // 0 or 8 (two adjacent float4s)
  const int br = tid >> 2;               // 0..63
  const int bc = (tid & 3) << 2;         // 0,4,8,12

  v8f acc[2][2] = {};

  float4 pa0, pa1, pb;
  // prologue: tile k0 = 0
  {
    const float* xr = X + (size_t)(mBase + ar) * IN_DIM;
    pa0 = *reinterpret_cast<const float4*>(xr + ac);
    pa1 = *reinterpret_cast<const float4*>(xr + ac + 4);
    pb  = *reinterpret_cast<const float4*>(W + (size_t)(nBase + br) * IN_DIM + bc);
    As[0][ar][ac + 0] = pa0.x; As[0][ar][ac + 1] = pa0.y;
    As[0][ar][ac + 2] = pa0.z; As[0][ar][ac + 3] = pa0.w;
    As[0][ar][ac + 4] = pa1.x; As[0][ar][ac + 5] = pa1.y;
    As[0][ar][ac + 6] = pa1.z; As[0][ar][ac + 7] = pa1.w;
    Bs[0][br][bc + 0] = pb.x;  Bs[0][br][bc + 1] = pb.y;
    Bs[0][br][bc + 2] = pb.z;  Bs[0][br][bc + 3] = pb.w;
  }
  __syncthreads();

  const int KT = IN_DIM / 16;   // 50
  for (int kt = 0; kt < KT; ++kt) {
    const int cur = kt & 1, nxt = cur ^ 1;
    if (kt + 1 < KT) {
      const int k0 = (kt + 1) * 16;
      const float* xr = X + (size_t)(mBase + ar) * IN_DIM + k0;
      pa0 = *reinterpret_cast<const float4*>(xr + ac);
      pa1 = *reinterpret_cast<const float4*>(xr + ac + 4);
      pb  = *reinterpret_cast<const float4*>(W + (size_t)(nBase + br) * IN_DIM + k0 + bc);
    }
    // compute on buffer `cur`
#pragma unroll
    for (int kb = 0; kb < 16; kb += 4) {
      v2f af[2], bf[2];
#pragma unroll
      for (int ti = 0; ti < 2; ++ti)
        af[ti] = ld2s(&As[cur][wm + ti * 16 + ln][kb + klo]);
#pragma unroll
      for (int tj = 0; tj < 2; ++tj)
        bf[tj] = ld2s(&Bs[cur][wn + tj * 16 + ln][kb + klo]);
#pragma unroll
      for (int ti = 0; ti < 2; ++ti)
#pragma unroll
        for (int tj = 0; tj < 2; ++tj)
          acc[ti][tj] = __builtin_amdgcn_wmma_f32_16x16x4_f32(
              false, af[ti], false, bf[tj], (short)0, acc[ti][tj], false, false);
    }
    if (kt + 1 < KT) {
      As[nxt][ar][ac + 0] = pa0.x; As[nxt][ar][ac + 1] = pa0.y;
      As[nxt][ar][ac + 2] = pa0.z; As[nxt][ar][ac + 3] = pa0.w;
      As[nxt][ar][ac + 4] = pa1.x; As[nxt][ar][ac + 5] = pa1.y;
      As[nxt][ar][ac + 6] = pa1.z; As[nxt][ar][ac + 7] = pa1.w;
      Bs[nxt][br][bc + 0] = pb.x;  Bs[nxt][br][bc + 1] = pb.y;
      Bs[nxt][br][bc + 2] = pb.z;  Bs[nxt][br][bc + 3] = pb.w;
    }
    __syncthreads();
  }

  // epilogue: add bias, store
#pragma unroll
  for (int tj = 0; tj < 2; ++tj) {
    const int col = nBase + wn + tj * 16 + ln;
    const float bn = bias[col];
#pragma unroll
    for (int ti = 0; ti < 2; ++ti) {
#pragma unroll
      for (int r = 0; r < 8; ++r) {
        const int row = mBase + wm + ti * 16 + r + hi8;
        out[(size_t)row * GATES + col] = acc[ti][tj][r] + bn;
      }
    }
  }
}

// ---------------------------------------------------------------------------
// Phase B: one timestep, both directions (blockIdx.z = dir).
//   gates(256x1600) = h_prev(256x400) @ W_hh^T + gatesPre[t]; fused LSTM cell.
// Block: 128 batch rows x 16 h-units x 4 gates. 8 waves; wave w = rows w*16..w*16+15.
// Each wave holds 4 accumulators (i,f,g,o) for its 16x16 (batch x unit) tile.
// ---------------------------------------------------------------------------
__global__ __launch_bounds__(256) void lstm_step(
    const float* __restrict__ gp0, const float* __restrict__ gp1,
    const float* __restrict__ whh_layer,   // [2][1600][400]
    const float* __restrict__ masks,       // [T][B]
    float* __restrict__ hbuf,              // [2 slots][2 dir][B*H]
    float* __restrict__ cbuf,              // [2 slots][2 dir][B*H]
    float* __restrict__ xout,              // [T][B][800]
    int s)
{
  __shared__ float As[2][128][LDPAD];      // h_prev tile
  __shared__ float Bs[2][64][LDPAD];       // W_hh rows: [gate*16 + unit][k]

  const int dir = blockIdx.z;
  const int t   = dir ? (T_SEQ - 1 - s) : s;
  const float* gp = dir ? gp1 : gp0;
  const float* W  = whh_layer + (size_t)dir * GATES * HID;

  const int prev = s & 1, next = prev ^ 1;
  const size_t dstate = (size_t)BATCH * HID;
  const float* hprev = hbuf + ((size_t)prev * 2 + dir) * dstate;
  float*       hnext = hbuf + ((size_t)next * 2 + dir) * dstate;
  const float* cprev = cbuf + ((size_t)prev * 2 + dir) * dstate;
  float*       cnext = cbuf + ((size_t)next * 2 + dir) * dstate;

  const int tid  = threadIdx.x;
  const int lane = tid & 31;
  const int wave = tid >> 5;
  const int ln   = lane & 15;
  const int klo  = (lane < 16) ? 0 : 2;
  const int hi8  = (lane < 16) ? 0 : 8;

  const int mBase = blockIdx.x * 128;
  const int u0    = blockIdx.y * 16;
  const int m0    = wave * 16;

  const int ar = tid >> 1;              // 0..127
  const int ac = (tid & 1) * 8;
  const int br = tid >> 2;              // 0..63: gate = br>>4, unit = br&15
  const int bc = (tid & 3) << 2;
  const int bg = br >> 4, bu = br & 15;

  v8f acc[4] = {};

  float4 pa0, pa1, pb;
  {
    const float* hr = hprev + (size_t)(mBase + ar) * HID;
    pa0 = *reinterpret_cast<const float4*>(hr + ac);
    pa1 = *reinterpret_cast<const float4*>(hr + ac + 4);
    pb  = *reinterpret_cast<const float4*>(W + ((size_t)bg * HID + u0 + bu) * HID + bc);
    As[0][ar][ac + 0] = pa0.x; As[0][ar][ac + 1] = pa0.y;
    As[0][ar][ac + 2] = pa0.z; As[0][ar][ac + 3] = pa0.w;
    As[0][ar][ac + 4] = pa1.x; As[0][ar][ac + 5] = pa1.y;
    As[0][ar][ac + 6] = pa1.z; As[0][ar][ac + 7] = pa1.w;
    Bs[0][br][bc + 0] = pb.x;  Bs[0][br][bc + 1] = pb.y;
    Bs[0][br][bc + 2] = pb.z;  Bs[0][br][bc + 3] = pb.w;
  }
  __syncthreads();

  const int KT = HID / 16;   // 25
  for (int kt = 0; kt < KT; ++kt) {
    const int cur = kt & 1, nxt = cur ^ 1;
    if (kt + 1 < KT) {
      const int k0 = (kt + 1) * 16;
      const float* hr = hprev + (size_t)(mBase + ar) * HID + k0;
      pa0 = *reinterpret_cast<const float4*>(hr + ac);
      pa1 = *reinterpret_cast<const float4*>(hr + ac + 4);
      pb  = *reinterpret_cast<const float4*>(W + ((size_t)bg * HID + u0 + bu) * HID + k0 + bc);
    }
#pragma unroll
    for (int kb = 0; kb < 16; kb += 4) {
      const v2f a = ld2s(&As[cur][m0 + ln][kb + klo]);
#pragma unroll
      for (int g = 0; g < 4; ++g) {
        const v2f b = ld2s(&Bs[cur][g * 16 + ln][kb + klo]);
        acc[g] = __builtin_amdgcn_wmma_f32_16x16x4_f32(
            false, a, false, b, (short)0, acc[g], false, false);
      }
    }
    if (kt + 1 < KT) {
      As[nxt][ar][ac + 0] = pa0.x; As[nxt][ar][ac + 1] = pa0.y;
      As[nxt][ar][ac + 2] = pa0.z; As[nxt][ar][ac + 3] = pa0.w;
      As[nxt][ar][ac + 4] = pa1.x; As[nxt][ar][ac + 5] = pa1.y;
      As[nxt][ar][ac + 6] = pa1.z; As[nxt][ar][ac + 7] = pa1.w;
      Bs[nxt][br][bc + 0] = pb.x;  Bs[nxt][br][bc + 1] = pb.y;
      Bs[nxt][br][bc + 2] = pb.z;  Bs[nxt][br][bc + 3] = pb.w;
    }
    __syncthreads();
  }

  // fused LSTM cell epilogue
  const int u = u0 + ln;
#pragma unroll
  for (int r = 0; r < 8; ++r) {
    const int m = mBase + m0 + r + hi8;
    const float* gpe = gp + ((size_t)t * BATCH + m) * GATES;
    const float xi = acc[0][r] + gpe[u];
    const float xf = acc[1][r] + gpe[HID + u];
    const float xg = acc[2][r] + gpe[2 * HID + u];
    const float xo = acc[3][r] + gpe[3 * HID + u];

    const float ig = fast_sigmoid(xi);
    const float fg = fast_sigmoid(xf);
    const float gg = fast_tanh(xg);
    const float og = fast_sigmoid(xo);

    const float cp   = cprev[(size_t)m * HID + u];
    const float craw = fg * cp + ig * gg;
    const float hraw = og * fast_tanh(craw);
    const float mt   = masks[(size_t)t * BATCH + m];
    const float hn   = hraw * mt;
    const float cn   = craw * mt;

    hnext[(size_t)m * HID + u] = hn;
    cnext[(size_t)m * HID + u] = cn;
    xout[((size_t)t * BATCH + m) * (2 * HID) + (size_t)dir * HID + u] = hn;
  }
}

// ---------------------------------------------------------------------------
// Host orchestration
// ---------------------------------------------------------------------------
extern "C" void kernel_launch(void* const* d_in, const int* in_sizes, int n_in,
                              void* d_out, int out_size, void* d_ws, size_t ws_size,
                              hipStream_t stream) {
  const float* x     = (const float*)d_in[0];   // (T,B,800)
  const float* masks = (const float*)d_in[1];   // (T,B,1)
  const float* w_ih  = (const float*)d_in[2];   // (L,2,1600,800)
  const float* w_hh  = (const float*)d_in[3];   // (L,2,1600,400)
  const float* bias  = (const float*)d_in[4];   // (L,2,1600)
  float* out = (float*)d_out;                   // (T,B,800)
  float* ws  = (float*)d_ws;

  const size_t TB = (size_t)T_SEQ * BATCH;      // 32768
  float* gp0  = ws;                             // TB*1600
  float* gp1  = gp0 + TB * GATES;               // TB*1600
  float* xbuf = gp1 + TB * GATES;               // TB*800
  float* hbuf = xbuf + TB * IN_DIM;             // 4 * B*H (2 slots x 2 dirs)
  float* cbuf = hbuf + (size_t)4 * BATCH * HID; // 4 * B*H

  const dim3 gridA(TB / 128, GATES / 64);       // 256 x 25
  const dim3 gridB(BATCH / 128, HID / 16, 2);   // 2 x 25 x 2

  for (int l = 0; l < LAYERS; ++l) {
    const float* xin = (l == 0) ? x : xbuf;
    float* xo = (l == LAYERS - 1) ? out : xbuf;   // xin fully consumed in phase A

    // Phase A: input-projection GEMMs for both directions
    gemm_gates<<<gridA, 256, 0, stream>>>(
        xin, w_ih + ((size_t)l * 2 + 0) * GATES * IN_DIM,
        bias + ((size_t)l * 2 + 0) * GATES, gp0);
    gemm_gates<<<gridA, 256, 0, stream>>>(
        xin, w_ih + ((size_t)l * 2 + 1) * GATES * IN_DIM,
        bias + ((size_t)l * 2 + 1) * GATES, gp1);

    // zero h0/c0 state (hbuf and cbuf are contiguous: 8*B*H floats)
    const int zn = 8 * BATCH * HID;
    zero_f32<<<(zn + 255) / 256, 256, 0, stream>>>(hbuf, zn);

    // Phase B: serial recurrence, both directions per launch
    for (int s = 0; s < T_SEQ; ++s) {
      lstm_step<<<gridB, 256, 0, stream>>>(
          gp0, gp1, w_hh + (size_t)l * 2 * GATES * HID,
          masks, hbuf, cbuf, xo, s);
    }
  }
}